// DegreeAwareEdgeEncoder_9294309228898
// MI455X (gfx1250) — compile-verified
//
#include <hip/hip_runtime.h>

// ---------------------------------------------------------------------------
// DegreeAwareEdgeEncoder for MI455X (gfx1250, wave32).
//
//   out[e, :] = du*W[0,:] + dv*W[1,:] + (du+dv)*W[2,:] + b
//
// HBM-bound: ~470 MB moved per call -> ~20us floor @ 23.3 TB/s. Compute is
// mapped onto V_WMMA_F32_16X16X4_F32: per 16-edge tile,
//   A(16x4) = [du, dv, du+dv, 1],  B(4x16 per N-tile) = [W0; W1; W2; b]
// so the bias rides in via the ones column and the matrix pipe does all FMAs.
// Output stores are non-temporal (write-once 410 MB stream > 192 MB L2),
// keeping L2 for the re-read edge list + degree counters.
// ---------------------------------------------------------------------------

typedef float v2f __attribute__((ext_vector_type(2)));
typedef float v8f __attribute__((ext_vector_type(8)));

// Interleaved degree counters: deg[2*node+0] = out-degree, deg[2*node+1] = in-degree.
// Interleaving means the host never needs to know num_nodes (device reads it).

__global__ void dae_zero_kernel(unsigned int* __restrict__ deg,
                                const int* __restrict__ num_nodes_p,
                                unsigned int ws_cap_u32) {
    const unsigned int n = (unsigned int)(*num_nodes_p);
    unsigned int total = 2u * n;
    if (total > ws_cap_u32) total = ws_cap_u32;
    const unsigned int stride = gridDim.x * blockDim.x;
    for (unsigned int i = blockIdx.x * blockDim.x + threadIdx.x; i < total; i += stride)
        deg[i] = 0u;
}

__global__ void dae_degree_kernel(const int* __restrict__ src,
                                  const int* __restrict__ dst,
                                  unsigned int* __restrict__ deg,
                                  int E) {
    const int stride = gridDim.x * blockDim.x;
    for (int e = blockIdx.x * blockDim.x + threadIdx.x; e < E; e += stride) {
        atomicAdd(&deg[2 * src[e] + 0], 1u);   // out-degree, no-return u32 atomic
        atomicAdd(&deg[2 * dst[e] + 1], 1u);   // in-degree
    }
}

__global__ void dae_encode_wmma_kernel(const int* __restrict__ src,
                                       const int* __restrict__ dst,
                                       const unsigned int* __restrict__ deg,
                                       const float* __restrict__ W,   // [3,32]
                                       const float* __restrict__ b,   // [32]
                                       float* __restrict__ out,       // [E,32]
                                       int E) {
    const int lane = threadIdx.x & 31;
    const int wave = threadIdx.x >> 5;
    const int wavesPerBlock = blockDim.x >> 5;
    const int tile = blockIdx.x * wavesPerBlock + wave;
    const int base = tile * 16;
    if (base >= E) return;                 // wave-uniform exit: EXEC stays all-1s

    const int m = lane & 15;               // edge-in-tile (A) / column (B, D)
    int e = base + m;
    if (e >= E) e = E - 1;                 // clamp for (non-occurring) tail
    const float du = (float)deg[2 * src[e] + 0];
    const float dv = (float)deg[2 * dst[e] + 1];

    // A-matrix 16x4 fp32 layout: lanes 0-15 -> K=0,1 ; lanes 16-31 -> K=2,3.
    v2f A;
    if (lane < 16) { A.x = du;      A.y = dv;   }   // K0=du, K1=dv
    else           { A.x = du + dv; A.y = 1.0f; }   // K2=du+dv, K3=1 (bias row)

    // B-matrix 4x16 fp32 layout: lane = column N; lanes 0-15 -> K=0,1 ; 16-31 -> K=2,3.
    const int c1 = m + 16;     // N-tile 1: emb cols 16..31
    v2f B0, B1;
    if (lane < 16) {
        B0.x = W[0 * 32 + m];  B0.y = W[1 * 32 + m];
        B1.x = W[0 * 32 + c1]; B1.y = W[1 * 32 + c1];
    } else {
        B0.x = W[2 * 32 + m];  B0.y = b[m];
        B1.x = W[2 * 32 + c1]; B1.y = b[c1];
    }

    v8f C0 = {};
    v8f C1 = {};
    // 8 args: (neg_a, A, neg_b, B, c_mod, C, reuse_a, reuse_b)
    C0 = __builtin_amdgcn_wmma_f32_16x16x4_f32(false, A, false, B0, (short)0, C0, false, false);
    C1 = __builtin_amdgcn_wmma_f32_16x16x4_f32(false, A, false, B1, (short)0, C1, false, false);

    // D layout: VGPR v -> row M=v (lanes 0-15) / M=v+8 (lanes 16-31), col N = lane&15.
    // Single per-lane base pointer; rows become 128-byte immediate offsets so the
    // 16 stores fold into one clause.
    const int rowOff = (lane < 16) ? 0 : 8;
    float* __restrict__ p = out + (size_t)(base + rowOff) * 32 + m;

    if (base + 16 <= E) {                  // wave-uniform fast path: full tile
#pragma unroll
        for (int v = 0; v < 8; ++v) {
            __builtin_nontemporal_store(C0[v], p + v * 32);        // cols 0..15
            __builtin_nontemporal_store(C1[v], p + v * 32 + 16);   // cols 16..31
        }
    } else {                               // tail tile (unused when E % 16 == 0)
#pragma unroll
        for (int v = 0; v < 8; ++v) {
            if (base + rowOff + v < E) {
                __builtin_nontemporal_store(C0[v], p + v * 32);
                __builtin_nontemporal_store(C1[v], p + v * 32 + 16);
            }
        }
    }
}

extern "C" void kernel_launch(void* const* d_in, const int* in_sizes, int n_in,
                              void* d_out, int out_size, void* d_ws, size_t ws_size,
                              hipStream_t stream) {
    const int*   edge_index  = (const int*)d_in[0];   // [2, E] flat
    const int*   num_nodes_p = (const int*)d_in[1];   // scalar on device
    const float* W           = (const float*)d_in[2]; // [3, 32]
    const float* b           = (const float*)d_in[3]; // [32]
    float*       out         = (float*)d_out;

    const int E = in_sizes[0] / 2;
    if (E <= 0) return;
    const int* src = edge_index;
    const int* dst = edge_index + E;

    unsigned int* deg = (unsigned int*)d_ws;
    const unsigned int ws_cap_u32 = (unsigned int)(ws_size / sizeof(unsigned int));

    // 1) Zero degree counters (device reads num_nodes; grid-stride covers 2*N).
    dae_zero_kernel<<<1024, 256, 0, stream>>>(deg, num_nodes_p, ws_cap_u32);

    // 2) Degree histograms via u32 atomics.
    {
        int blocks = (E + 255) / 256;
        if (blocks > 65535) blocks = 65535;   // grid-stride handles the rest
        dae_degree_kernel<<<blocks, 256, 0, stream>>>(src, dst, deg, E);
    }

    // 3) WMMA encode: one wave per 16-edge tile, 8 waves per 256-thread block.
    {
        const int numTiles = (E + 15) / 16;
        const int wavesPerBlock = 8;
        const int blocks = (numTiles + wavesPerBlock - 1) / wavesPerBlock;
        dae_encode_wmma_kernel<<<blocks, 32 * wavesPerBlock, 0, stream>>>(
            src, dst, deg, W, b, out, E);
    }
}